// Network_44968307589213
// MI455X (gfx1250) — compile-verified
//
#include <hip/hip_runtime.h>
#include <math.h>

typedef float v2f __attribute__((ext_vector_type(2)));
typedef float v8f __attribute__((ext_vector_type(8)));

#define BATCH 2
#define NPTS  32768
#define KNBR  16
#define DIN   32
#define DOUT  64
#define HCH   32          // DOUT/2
#define WAVES 8           // waves per block (256 threads)

__device__ __forceinline__ float bn_scale(float g) { return g * rsqrtf(1.0f + 1e-5f); }

__device__ __forceinline__ float red_max16(float v) {
#pragma unroll
  for (int m = 1; m < 16; m <<= 1) v = fmaxf(v, __shfl_xor(v, m, 16));
  return v;
}
__device__ __forceinline__ float red_sum16(float v) {
#pragma unroll
  for (int m = 1; m < 16; m <<= 1) v += __shfl_xor(v, m, 16);
  return v;
}

// ---------------- Kernel 1: f_pc = relu(bn(w_mlp1 @ feature)) -> (B*N, 32) ----
__global__ void __launch_bounds__(256)
k1_fpc(const float* __restrict__ feature, const float* __restrict__ w,
       const float* __restrict__ g, const float* __restrict__ bb,
       float* __restrict__ fpc) {
  int tid = blockIdx.x * 256 + threadIdx.x;
  int p = tid >> 5;          // point id over B*N  (32 channels per point)
  int c = tid & 31;
  if (p >= BATCH * NPTS) return;
  int b = p >> 15, n = p & (NPTS - 1);
  const float* f = feature + ((size_t)b * DIN) * NPTS + n;
  float acc = 0.f;
#pragma unroll
  for (int j = 0; j < DIN; ++j) acc += w[c * DIN + j] * f[(size_t)j * NPTS];
  float v = acc * bn_scale(g[c]) + bb[c];
  fpc[(size_t)p * HCH + c] = fmaxf(v, 0.f);
}

// ---- helper: rel-pos encoding + bb_w1 conv/bn/relu for lane's neighbor k ----
// lanes 0..15 only; writes results into fx1[0..31] (registers)
__device__ __forceinline__ void relpos_bb1(const float* __restrict__ xyz,
                                           const int* __restrict__ nidx,
                                           const float* __restrict__ w1,
                                           const float* __restrict__ g1,
                                           const float* __restrict__ b1,
                                           int b, int n, int k, float fx1[HCH]) {
  int idx = nidx[((size_t)(b * NPTS + n)) * KNBR + k];
  const float* pc = xyz + ((size_t)(b * NPTS + n)) * 3;
  const float* qc = xyz + ((size_t)(b * NPTS + idx)) * 3;
  float px = pc[0], py = pc[1], pz = pc[2];
  float qx = qc[0], qy = qc[1], qz = qc[2];
  float rx = px - qx, ry = py - qy, rz = pz - qz;
  float dist = sqrtf(rx * rx + ry * ry + rz * rz);
  float f10[10] = {dist, rx, ry, rz, px, py, pz, qx, qy, qz};
#pragma unroll 4
  for (int c = 0; c < HCH; ++c) {
    float s = 0.f;
#pragma unroll
    for (int j = 0; j < 10; ++j) s += w1[c * 10 + j] * f10[j];
    float v = s * bn_scale(g1[c]) + b1[c];
    fx1[c] = fmaxf(v, 0.f);
  }
}

// ---- helper: score = fc(64x64) @ feat(64x16) via WMMA f32 16x16x4,
//      softmax over neighbor columns, agg[row] = sum_k feat*score -> LDS agg ---
__device__ __forceinline__ void att_score_agg(const float* __restrict__ fcL,
                                              float (*feat)[KNBR],
                                              float* __restrict__ agg,
                                              int lo, int hi) {
  v8f acc[4] = {v8f{}, v8f{}, v8f{}, v8f{}};
#pragma unroll
  for (int kt = 0; kt < 16; ++kt) {
    const int k0 = kt * 4;
    v2f bv;                                   // B: K x N tile (4x16)
    bv.x = feat[k0 + 2 * hi + 0][lo];
    bv.y = feat[k0 + 2 * hi + 1][lo];
#pragma unroll
    for (int t = 0; t < 4; ++t) {             // A: M x K tile (16x4) from fc
      v2f av;
      av.x = fcL[(t * 16 + lo) * DOUT + k0 + 2 * hi + 0];
      av.y = fcL[(t * 16 + lo) * DOUT + k0 + 2 * hi + 1];
      acc[t] = __builtin_amdgcn_wmma_f32_16x16x4_f32(
          false, av, false, bv, (short)0, acc[t], false, false);
    }
  }
  // softmax over 16 neighbor columns (the 16-lane half), then weighted sum
#pragma unroll
  for (int t = 0; t < 4; ++t) {
#pragma unroll
    for (int i = 0; i < 8; ++i) {
      float v  = acc[t][i];
      float mx = red_max16(v);
      float e  = __expf(v - mx);
      float sm = red_sum16(e);
      float sc = e / sm;
      int row  = t * 16 + i + 8 * hi;         // C/D layout row
      float p  = feat[row][lo] * sc;
      float a  = red_sum16(p);
      if (lo == 0) agg[row] = a;              // lane0 of each half writes
    }
  }
  asm volatile("s_wait_dscnt 0" ::: "memory");
}

// ---------------- Kernel 2: stage-1 attentive pool -> f_agg1 (B*N, 32) -------
__global__ void __launch_bounds__(256)
k2_stage1(const float* __restrict__ xyz, const int* __restrict__ nidx,
          const float* __restrict__ fpc,
          const float* __restrict__ bb_w1, const float* __restrict__ bb_g1,
          const float* __restrict__ bb_b1, const float* __restrict__ att_fc,
          const float* __restrict__ att_w, const float* __restrict__ att_g,
          const float* __restrict__ att_b, float* __restrict__ fagg1) {
  __shared__ float fcL[DOUT * DOUT];
  __shared__ float featL[WAVES][DOUT][KNBR];
  __shared__ float aggL[WAVES][DOUT];

  for (int i = threadIdx.x; i < DOUT * DOUT; i += 256) fcL[i] = att_fc[i];
  __syncthreads();

  const int lane = threadIdx.x & 31, wid = threadIdx.x >> 5;
  const int lo = lane & 15, hi = lane >> 4;
  const int pid = blockIdx.x * WAVES + wid;
  const int b = pid >> 15, n = pid & (NPTS - 1);
  float (*feat)[KNBR] = featL[wid];
  float* agg = aggL[wid];

  if (hi == 0) {  // lanes 0..15: one neighbor each
    float fx1[HCH];
    relpos_bb1(xyz, nidx, bb_w1, bb_g1, bb_b1, b, n, lo, fx1);
#pragma unroll
    for (int c = 0; c < HCH; ++c) feat[HCH + c][lo] = fx1[c];  // rows 32..63
    int idx = nidx[((size_t)(b * NPTS + n)) * KNBR + lo];
    const float4* fp4 = (const float4*)(fpc + (size_t)(b * NPTS + idx) * HCH);
#pragma unroll
    for (int t = 0; t < 8; ++t) {                              // rows 0..31
      float4 v = fp4[t];
      feat[4 * t + 0][lo] = v.x; feat[4 * t + 1][lo] = v.y;
      feat[4 * t + 2][lo] = v.z; feat[4 * t + 3][lo] = v.w;
    }
  }
  asm volatile("s_wait_dscnt 0" ::: "memory");

  att_score_agg(fcL, feat, agg, lo, hi);

  // mlp: 32 outputs, one per lane
  {
    int o = lane;
    float s = 0.f;
#pragma unroll 8
    for (int c = 0; c < DOUT; ++c) s += att_w[o * DOUT + c] * agg[c];
    float v = s * bn_scale(att_g[o]) + att_b[o];
    fagg1[(size_t)pid * HCH + o] = fmaxf(v, 0.f);
  }
}

// ---------------- Kernel 3: stage-2 + mlp2 + shortcut + leaky relu -----------
__global__ void __launch_bounds__(256)
k3_stage2(const float* __restrict__ feature, const float* __restrict__ xyz,
          const int* __restrict__ nidx, const float* __restrict__ fagg1,
          const float* __restrict__ bb_w1, const float* __restrict__ bb_g1,
          const float* __restrict__ bb_b1, const float* __restrict__ bb_w2,
          const float* __restrict__ bb_g2, const float* __restrict__ bb_b2,
          const float* __restrict__ att_fc, const float* __restrict__ att_w,
          const float* __restrict__ att_g, const float* __restrict__ att_b,
          const float* __restrict__ w_mlp2, const float* __restrict__ g_mlp2,
          const float* __restrict__ b_mlp2, const float* __restrict__ w_sc,
          const float* __restrict__ g_sc, const float* __restrict__ b_sc,
          float* __restrict__ out) {
  __shared__ float fcL[DOUT * DOUT];
  __shared__ float featL[WAVES][DOUT][KNBR];
  __shared__ float aggL[WAVES][DOUT];
  __shared__ float agg2L[WAVES][DOUT];
  __shared__ float finL[WAVES][DIN];

  for (int i = threadIdx.x; i < DOUT * DOUT; i += 256) fcL[i] = att_fc[i];
  __syncthreads();

  const int lane = threadIdx.x & 31, wid = threadIdx.x >> 5;
  const int lo = lane & 15, hi = lane >> 4;
  const int pid = blockIdx.x * WAVES + wid;
  const int b = pid >> 15, n = pid & (NPTS - 1);
  float (*feat)[KNBR] = featL[wid];
  float* agg = aggL[wid];
  float* agg2 = agg2L[wid];
  float* fin = finL[wid];

  // point's input feature vector -> LDS (for shortcut)
  fin[lane] = feature[((size_t)b * DIN + lane) * NPTS + n];

  if (hi == 0) {  // lanes 0..15
    float fx1[HCH];
    relpos_bb1(xyz, nidx, bb_w1, bb_g1, bb_b1, b, n, lo, fx1);
    // f_xyz stage 2: bb_w2 (32x32) + bn + relu -> rows 32..63
#pragma unroll 4
    for (int c2 = 0; c2 < HCH; ++c2) {
      float s = 0.f;
#pragma unroll
      for (int c = 0; c < HCH; ++c) s += bb_w2[c2 * HCH + c] * fx1[c];
      float v = s * bn_scale(bb_g2[c2]) + bb_b2[c2];
      feat[HCH + c2][lo] = fmaxf(v, 0.f);
    }
    int idx = nidx[((size_t)(b * NPTS + n)) * KNBR + lo];
    const float4* fa4 = (const float4*)(fagg1 + (size_t)(b * NPTS + idx) * HCH);
#pragma unroll
    for (int t = 0; t < 8; ++t) {                              // rows 0..31
      float4 v = fa4[t];
      feat[4 * t + 0][lo] = v.x; feat[4 * t + 1][lo] = v.y;
      feat[4 * t + 2][lo] = v.z; feat[4 * t + 3][lo] = v.w;
    }
  }
  asm volatile("s_wait_dscnt 0" ::: "memory");

  att_score_agg(fcL, feat, agg, lo, hi);

  // att2 mlp: 64 outputs, two per lane -> agg2
#pragma unroll
  for (int q = 0; q < 2; ++q) {
    int o = lane + 32 * q;
    float s = 0.f;
#pragma unroll 8
    for (int c = 0; c < DOUT; ++c) s += att_w[o * DOUT + c] * agg[c];
    float v = s * bn_scale(att_g[o]) + att_b[o];
    agg2[o] = fmaxf(v, 0.f);
  }
  asm volatile("s_wait_dscnt 0" ::: "memory");

  // mlp2 (128x64) + shortcut (128x32) + leaky relu, four outputs per lane
#pragma unroll
  for (int q = 0; q < 4; ++q) {
    int o = lane + 32 * q;
    float s = 0.f;
#pragma unroll 8
    for (int c = 0; c < DOUT; ++c) s += w_mlp2[o * DOUT + c] * agg2[c];
    float fo = s * bn_scale(g_mlp2[o]) + b_mlp2[o];
    float ss = 0.f;
#pragma unroll 8
    for (int c = 0; c < DIN; ++c) ss += w_sc[o * DIN + c] * fin[c];
    ss = ss * bn_scale(g_sc[o]) + b_sc[o];
    float y = fo + ss;
    y = (y > 0.f) ? y : 0.2f * y;
    out[((size_t)b * 2 * DOUT + o) * NPTS + n] = y;
  }
}

extern "C" void kernel_launch(void* const* d_in, const int* in_sizes, int n_in,
                              void* d_out, int out_size, void* d_ws, size_t ws_size,
                              hipStream_t stream) {
  const float* feature = (const float*)d_in[0];
  const float* xyz     = (const float*)d_in[1];
  const float* w_mlp1  = (const float*)d_in[2];
  const float* g_mlp1  = (const float*)d_in[3];
  const float* b_mlp1  = (const float*)d_in[4];
  const float* bb_w1   = (const float*)d_in[5];
  const float* bb_g1   = (const float*)d_in[6];
  const float* bb_b1   = (const float*)d_in[7];
  const float* att1_fc = (const float*)d_in[8];
  const float* att1_w  = (const float*)d_in[9];
  const float* att1_g  = (const float*)d_in[10];
  const float* att1_b  = (const float*)d_in[11];
  const float* bb_w2   = (const float*)d_in[12];
  const float* bb_g2   = (const float*)d_in[13];
  const float* bb_b2   = (const float*)d_in[14];
  const float* att2_fc = (const float*)d_in[15];
  const float* att2_w  = (const float*)d_in[16];
  const float* att2_g  = (const float*)d_in[17];
  const float* att2_b  = (const float*)d_in[18];
  const float* w_mlp2  = (const float*)d_in[19];
  const float* g_mlp2  = (const float*)d_in[20];
  const float* b_mlp2  = (const float*)d_in[21];
  const float* w_sc    = (const float*)d_in[22];
  const float* g_sc    = (const float*)d_in[23];
  const float* b_sc    = (const float*)d_in[24];
  const int*   nidx    = (const int*)d_in[25];
  float* out = (float*)d_out;

  float* fpc   = (float*)d_ws;                         // (B*N, 32)
  float* fagg1 = fpc + (size_t)BATCH * NPTS * HCH;     // (B*N, 32)

  const int pts_blocks = (BATCH * NPTS) / WAVES;       // 8192

  k1_fpc<<<(BATCH * NPTS * HCH) / 256, 256, 0, stream>>>(
      feature, w_mlp1, g_mlp1, b_mlp1, fpc);

  k2_stage1<<<pts_blocks, 256, 0, stream>>>(
      xyz, nidx, fpc, bb_w1, bb_g1, bb_b1,
      att1_fc, att1_w, att1_g, att1_b, fagg1);

  k3_stage2<<<pts_blocks, 256, 0, stream>>>(
      feature, xyz, nidx, fagg1, bb_w1, bb_g1, bb_b1, bb_w2, bb_g2, bb_b2,
      att2_fc, att2_w, att2_g, att2_b, w_mlp2, g_mlp2, b_mlp2,
      w_sc, g_sc, b_sc, out);
}